// CAFE_Interpolation_42717744726573
// MI455X (gfx1250) — compile-verified
//
#include <hip/hip_runtime.h>
#include <hip/hip_bf16.h>
#include <stdint.h>

typedef __attribute__((ext_vector_type(2))) float v2f;
typedef __attribute__((ext_vector_type(4))) float v4f;
typedef __attribute__((ext_vector_type(8))) float v8f;

#define B_ 128
#define T_ 1024
#define D_ 512

// ---------------------------------------------------------------------------
// Phase 1: im[b,d] = (1/T) * sum_t grad[b,t,d] * x[b,t,d]
// Implemented as 1^T @ (G .* X) using V_WMMA_F32_16X16X4_F32 (exact fp32 FMA).
// One wave handles one (batch, 16-column d-group); 4096 waves total.
// A = ones(16x4)  =>  D[m,n] = sum_k B[k,n] for every m, so each lane's c[0]
// is the running column sum for column n = lane%16 (layout-robust).
// grad is streamed with NT loads; x uses RT so it can stay hot in the 192MB L2
// for phase 3.
// ---------------------------------------------------------------------------
__global__ __launch_bounds__(256) void cafe_phase1_im(
    const float* __restrict__ x, const float* __restrict__ g,
    float* __restrict__ im)
{
  const int wave = blockIdx.x * (blockDim.x >> 5) + (threadIdx.x >> 5);
  const int lane = threadIdx.x & 31;
  const int b    = wave >> 5;   // 32 d-groups of 16 per batch
  const int dg   = wave & 31;
  const int col  = lane & 15;
  const int half = (lane >> 4) & 1;          // selects t-row pair within K=4
  const size_t d = (size_t)dg * 16 + col;
  const size_t rowbase = (size_t)b * T_;

  v2f a; a[0] = 1.0f; a[1] = 1.0f;           // A matrix: all ones
  v8f c0 = {}, c1 = {}, c2 = {}, c3 = {};    // 4 accumulators: break WMMA RAW chain

#define CAFE_ACC(C, TT)                                                        \
  {                                                                            \
    size_t base = (rowbase + (size_t)(t0 + (TT) + 2 * half)) * D_ + d;         \
    float x0 = x[base];                                                        \
    float x1 = x[base + D_];                                                   \
    float g0 = __builtin_nontemporal_load(&g[base]);                           \
    float g1 = __builtin_nontemporal_load(&g[base + D_]);                      \
    v2f bm; bm[0] = x0 * g0; bm[1] = x1 * g1;                                  \
    C = __builtin_amdgcn_wmma_f32_16x16x4_f32(false, a, false, bm, (short)0,   \
                                              C, false, false);                \
  }

  for (int t0 = 0; t0 < T_; t0 += 16) {
    CAFE_ACC(c0, 0)
    CAFE_ACC(c1, 4)
    CAFE_ACC(c2, 8)
    CAFE_ACC(c3, 12)
  }
#undef CAFE_ACC

  float s = (c0[0] + c1[0]) + (c2[0] + c3[0]);
  if (lane < 16)
    im[(size_t)b * D_ + dg * 16 + lane] = s * (1.0f / (float)T_);
}

// ---------------------------------------------------------------------------
// Phase 2: per-batch 0.9-quantile over D=512 values + packed mask bits.
// Rank-by-counting with index tie-break gives exact order statistics.
// thr = v[459] + 0.9*(v[460]-v[459])  (numpy/jnp 'linear' method, q*(n-1)=459.9)
// mask bit d = (im[b,d] > thr), packed 32 bits/word via wave32 ballot.
// ---------------------------------------------------------------------------
__global__ __launch_bounds__(512) void cafe_phase2_thr(
    const float* __restrict__ im, unsigned* __restrict__ maskw)
{
  __shared__ float vals[D_];
  __shared__ float vlo, vhi;
  const int b = blockIdx.x;
  const int d = threadIdx.x;

  float v = im[(size_t)b * D_ + d];
  vals[d] = v;
  __syncthreads();

  int rank = 0;
#pragma unroll 8
  for (int j = 0; j < D_; ++j) {
    float o = vals[j];
    rank += (int)((o < v) || (o == v && j < d));
  }
  if (rank == 459) vlo = v;
  if (rank == 460) vhi = v;
  __syncthreads();

  float thr = vlo + 0.9f * (vhi - vlo);
  unsigned long long bal = __ballot(v > thr);   // wave32: low 32 bits valid
  if ((d & 31) == 0) maskw[b * 16 + (d >> 5)] = (unsigned)bal;
}

// ---------------------------------------------------------------------------
// Phase 3: out[b,t,d]:
//   non-dominant: sg + sr == x exactly (mask is 0/1)
//   dominant:     x*(1-mb) + (m*(x*mb) + (1-m)*(xp*mb_p))  -- literal ref order
// v4f (b128) loads/stores; NT store for out (never re-read) so x can stay
// in L2; tiny per-batch tables (mix/partner/is_dom/masks) hit WGP$/L2.
// ---------------------------------------------------------------------------
__global__ __launch_bounds__(256) void cafe_phase3_out(
    const float* __restrict__ x, const float* __restrict__ mix,
    const int* __restrict__ partner, const unsigned char* __restrict__ is_dom,
    const unsigned* __restrict__ maskw, float* __restrict__ out)
{
  const int i  = blockIdx.x * blockDim.x + threadIdx.x;   // over B*T*D/4
  const int d4 = i & (D_ / 4 - 1);
  const int bt = i >> 7;            // D_/4 == 128
  const int b  = bt >> 10;          // T_ == 1024
  const int t  = bt & (T_ - 1);
  const int d  = d4 << 2;

  const v4f* x4 = (const v4f*)x;
  v4f xv = x4[i];
  v4f o;

  if (is_dom[b]) {
    float m  = mix[b];
    float om = 1.0f - m;
    int   p  = partner[b];
    size_t pi = ((size_t)p * T_ + t) * (D_ / 4) + d4;
    v4f xp = x4[pi];
    unsigned mb  = (maskw[b * 16 + (d >> 5)] >> (d & 31)) & 0xFu;
    unsigned mp  = (maskw[p * 16 + (d >> 5)] >> (d & 31)) & 0xFu;
#pragma unroll
    for (int j = 0; j < 4; ++j) {
      unsigned bit = 1u << j;
      float sg  = (mb & bit) ? 0.0f : xv[j];
      float sr  = (mb & bit) ? xv[j] : 0.0f;
      float srp = (mp & bit) ? xp[j] : 0.0f;
      o[j] = sg + (m * sr + om * srp);
    }
  } else {
    o = xv;   // sg + sr == x exactly
  }
  __builtin_nontemporal_store(o, &((v4f*)out)[i]);
}

// ---------------------------------------------------------------------------
// Inputs (setup_inputs order):
//   0: x (B,T,D) f32   1: scenario_gradient (B,T,D) f32   2: mixup_strength (B) f32
//   3: scenario (B) i64 (unused)   4: partner_idx (B) i32   5: is_dominant (B) bool
// Workspace: im = B*D f32 (256KB) @ 0 ; mask = B*16 u32 (8KB) after it.
// ---------------------------------------------------------------------------
extern "C" void kernel_launch(void* const* d_in, const int* in_sizes, int n_in,
                              void* d_out, int out_size, void* d_ws, size_t ws_size,
                              hipStream_t stream)
{
  (void)in_sizes; (void)n_in; (void)out_size; (void)ws_size;
  const float*         xx      = (const float*)d_in[0];
  const float*         gg      = (const float*)d_in[1];
  const float*         mix     = (const float*)d_in[2];
  const int*           partner = (const int*)d_in[4];
  const unsigned char* is_dom  = (const unsigned char*)d_in[5];
  float*               out     = (float*)d_out;

  float*    ws_im   = (float*)d_ws;
  unsigned* ws_mask = (unsigned*)((char*)d_ws + (size_t)B_ * D_ * sizeof(float));

  // Phase 1: 4096 waves = 512 blocks x 8 waves (wave32)
  cafe_phase1_im<<<512, 256, 0, stream>>>(xx, gg, ws_im);
  // Phase 2: one 512-thread block per batch
  cafe_phase2_thr<<<B_, D_, 0, stream>>>(ws_im, ws_mask);
  // Phase 3: B*T*D/4 float4 threads
  cafe_phase3_out<<<(B_ * T_ * (D_ / 4)) / 256, 256, 0, stream>>>(
      xx, mix, partner, is_dom, ws_mask, out);
}